// DIDP_30760555774210
// MI455X (gfx1250) — compile-verified
//
#include <hip/hip_runtime.h>

#define N_NODE 500000
#define EMB 64
#define NNZ 4000000
#define BATCH 64
#define SEQ 200
#define EPS_F 1e-12f

typedef __attribute__((ext_vector_type(2))) float v2f;
typedef __attribute__((ext_vector_type(8))) float v8f;

// ---------------------------------------------------------------------------
// SpMM scatter: y[row,:] += val * x[col,:]  (COO, 16 lanes per edge, float4)
// Gathers are coalesced 256B rows; atomics hit L2 (dest fits in 192MB L2).
// ---------------------------------------------------------------------------
__global__ __launch_bounds__(256) void spmm_scatter(
    const float* __restrict__ x, const float* __restrict__ vals,
    const int* __restrict__ rows, const int* __restrict__ cols,
    float* __restrict__ y)
{
    int t = blockIdx.x * blockDim.x + threadIdx.x;
    int e = t >> 4;                       // 16 threads per edge
    if (e >= NNZ) return;
    int sub = t & 15;                     // which float4 of the 64-float row

    int   c = cols[e];
    int   r = rows[e];
    float v = vals[e];

    const float4 xv = *reinterpret_cast<const float4*>(x + (size_t)c * EMB + sub * 4);
    float* dst = y + (size_t)r * EMB + sub * 4;
    atomicAdd(dst + 0, xv.x * v);
    atomicAdd(dst + 1, xv.y * v);
    atomicAdd(dst + 2, xv.z * v);
    atomicAdd(dst + 3, xv.w * v);
}

// ---------------------------------------------------------------------------
// Fused gather + L2-normalize + sum:
//   out[i,:] = emb[s] + u1[s]/max(||u1[s]||,eps) + u2[s]/max(||u2[s]||,eps)
// One wave (32 lanes) per 16 gathered rows. Row sums-of-squares are computed
// with V_WMMA_F32_16X16X4_F32:  D = (X∘X) * ones  ->  D[m,n] = ||row m||^2.
// A-tile layout (32-bit A 16x4): lanes 0-15 supply K={0,1}, lanes 16-31 K={2,3}.
// C/D layout: VGPR v -> M=v (lanes 0-15) / M=v+8 (lanes 16-31).
// ---------------------------------------------------------------------------
__global__ __launch_bounds__(32) void gather_norm_out(
    const float* __restrict__ ue, const float* __restrict__ u1,
    const float* __restrict__ u2, const int* __restrict__ seq,
    float* __restrict__ out)
{
    const int lane  = threadIdx.x;
    const int base  = blockIdx.x * 16;       // 16 seq entries per wave
    const int m     = lane & 15;             // row this lane feeds into A
    const int node  = seq[base + m];
    const size_t roff = (size_t)node * EMB;
    const int kbase = (lane >> 4) * 2;       // 0 for lanes<16, 2 for lanes>=16

    v2f b; b[0] = 1.0f; b[1] = 1.0f;         // all-ones B (4x16) -> row reduction
    v8f c1 = {}; v8f c2 = {};

    for (int k0 = 0; k0 < EMB; k0 += 4) {
        float2 x1 = *reinterpret_cast<const float2*>(u1 + roff + k0 + kbase);
        float2 x2 = *reinterpret_cast<const float2*>(u2 + roff + k0 + kbase);
        v2f a1; a1[0] = x1.x * x1.x; a1[1] = x1.y * x1.y;
        v2f a2; a2[0] = x2.x * x2.x; a2[1] = x2.y * x2.y;
        // 8 args: (neg_a, A, neg_b, B, c_mod, C, reuse_a, reuse_b)
        c1 = __builtin_amdgcn_wmma_f32_16x16x4_f32(false, a1, false, b,
                                                   (short)0, c1, false, false);
        c2 = __builtin_amdgcn_wmma_f32_16x16x4_f32(false, a2, false, b,
                                                   (short)0, c2, false, false);
    }

    // c1[v]: lanes 0-15 hold ||row v||^2, lanes 16-31 hold ||row v+8||^2
    for (int r = 0; r < 16; ++r) {
        int   src  = (r < 8) ? 0 : 16;
        float n2_1 = __shfl(c1[r & 7], src, 32);
        float n2_2 = __shfl(c2[r & 7], src, 32);
        float inv1 = 1.0f / fmaxf(sqrtf(n2_1), EPS_F);
        float inv2 = 1.0f / fmaxf(sqrtf(n2_2), EPS_F);
        int   nd   = __shfl(node, r, 32);

        size_t ro = (size_t)nd * EMB + lane * 2;          // coalesced float2
        float2 e0 = *reinterpret_cast<const float2*>(ue + ro);
        float2 a0 = *reinterpret_cast<const float2*>(u1 + ro);
        float2 b0 = *reinterpret_cast<const float2*>(u2 + ro);
        float2 o;
        o.x = e0.x + a0.x * inv1 + b0.x * inv2;
        o.y = e0.y + a0.y * inv1 + b0.y * inv2;
        *reinterpret_cast<float2*>(out + (size_t)(base + r) * EMB + lane * 2) = o;
    }
}

extern "C" void kernel_launch(void* const* d_in, const int* in_sizes, int n_in,
                              void* d_out, int out_size, void* d_ws, size_t ws_size,
                              hipStream_t stream) {
    (void)in_sizes; (void)n_in; (void)out_size; (void)ws_size;
    const float* user_emb = (const float*)d_in[0];
    const float* h_values = (const float*)d_in[1];
    const int*   h_rows   = (const int*)d_in[2];
    const int*   h_cols   = (const int*)d_in[3];
    const int*   seq      = (const int*)d_in[4];
    float*       out      = (float*)d_out;

    const size_t ebytes = (size_t)N_NODE * EMB * sizeof(float);   // 128 MB
    float* u1 = (float*)d_ws;
    float* u2 = (float*)((char*)d_ws + ebytes);

    hipMemsetAsync(u1, 0, ebytes, stream);
    hipMemsetAsync(u2, 0, ebytes, stream);

    const int threads = 256;
    const int blocks  = (NNZ * 16 + threads - 1) / threads;       // 250000

    // u1 = spmm(emb); u2 = spmm(u1)   (reference propagates the RAW state)
    spmm_scatter<<<blocks, threads, 0, stream>>>(user_emb, h_values, h_rows, h_cols, u1);
    spmm_scatter<<<blocks, threads, 0, stream>>>(u1,       h_values, h_rows, h_cols, u2);

    // out[seq] = emb + normalize(u1) + normalize(u2), fused at the gather
    gather_norm_out<<<(BATCH * SEQ) / 16, 32, 0, stream>>>(user_emb, u1, u2, seq, out);
}